// ATTHScore_8684423872521
// MI455X (gfx1250) — compile-verified
//
#include <hip/hip_runtime.h>
#include <hip/hip_bf16.h>

typedef __attribute__((ext_vector_type(2))) float v2f;
typedef __attribute__((ext_vector_type(8))) float v8f;

#define MIN_NORM 1e-15f
#define BALL_EPS 1e-5f

__device__ __forceinline__ float wred(float v) {
    #pragma unroll
    for (int o = 16; o > 0; o >>= 1) v += __shfl_xor(v, o, 32);
    return v;
}

__device__ __forceinline__ float tanh_clip(float x) {
    x = fminf(fmaxf(x, -15.f), 15.f);
    float e = __expf(2.f * x);
    return (e - 1.f) / (e + 1.f);
}

// LDS byte-offset of a pointer into a __shared__ array (addrspace(3) value).
__device__ __forceinline__ unsigned lds_off(const void* p) {
    return (unsigned)(unsigned long long)(__attribute__((address_space(3))) const char*)p;
}

// CDNA5 async global->LDS copy, 16B per lane, tracked by ASYNCcnt.
// saddr form: 64-bit SGPR base + signed 32-bit per-lane byte offset.
__device__ __forceinline__ void async_copy_b128(unsigned lds_addr, const float* sbase,
                                                unsigned byte_off) {
    asm volatile("global_load_async_to_lds_b128 %0, %1, %2"
                 :: "v"(lds_addr), "v"(byte_off), "s"(sbase) : "memory");
}

__device__ __forceinline__ void wait_async0() {
    asm volatile("s_wait_asynccnt 0x0" ::: "memory");
}

// ---------------------------------------------------------------------------
// Kernel 1: per-row geometry. One wave32 per row of B=16384, d=512.
// Lane j-chunk mapping: chunk j covers elements [j*64 + lane*2, +1] (one
// rotation pair per float2, fully coalesced 256B wave segments).
// Writes res row (f32) and rowpack = {c, sqrt_c, x2, head_bias}.
// ---------------------------------------------------------------------------
__global__ __launch_bounds__(256) void atth_prep(
    const float* __restrict__ head, const float* __restrict__ head_bias,
    const float* __restrict__ rel, const float* __restrict__ rel_diag,
    const float* __restrict__ curv, const float* __restrict__ context,
    const float* __restrict__ scale,
    float* __restrict__ res, float4* __restrict__ rowpack)
{
    const int lane = threadIdx.x & 31;
    const int row  = blockIdx.x * 8 + (threadIdx.x >> 5);
    const float* hp   = head     + (size_t)row * 512;
    const float* rotp = rel_diag + (size_t)row * 1024;
    const float* refp = rotp + 512;
    const float* ctxp = context  + (size_t)row * 512;
    const float* relp = rel      + (size_t)row * 512;
    const float s = scale[0];

    float2 hx[8], rotq[8], refq[8];
    float dr = 0.f, df = 0.f;
    #pragma unroll
    for (int j = 0; j < 8; j++) {
        const int idx = j * 64 + lane * 2;
        hx[j]     = *(const float2*)(hp + idx);
        float2 rg = *(const float2*)(rotp + idx);
        float2 fg = *(const float2*)(refp + idx);
        float2 cx = *(const float2*)(ctxp + idx);
        // rotation: g = rg/|rg| ; out = (g0*x0 - g1*x1, g0*x1 + g1*x0)
        float inv = 1.f / fmaxf(sqrtf(rg.x * rg.x + rg.y * rg.y), MIN_NORM);
        float g0 = rg.x * inv, g1 = rg.y * inv;
        rotq[j].x = g0 * hx[j].x - g1 * hx[j].y;
        rotq[j].y = g0 * hx[j].y + g1 * hx[j].x;
        // reflection: out = (g0*x0 + g1*x1, -g0*x1 + g1*x0)
        inv = 1.f / fmaxf(sqrtf(fg.x * fg.x + fg.y * fg.y), MIN_NORM);
        g0 = fg.x * inv; g1 = fg.y * inv;
        refq[j].x =  g0 * hx[j].x + g1 * hx[j].y;
        refq[j].y = -g0 * hx[j].y + g1 * hx[j].x;
        dr += cx.x * rotq[j].x + cx.y * rotq[j].y;
        df += cx.x * refq[j].x + cx.y * refq[j].y;
    }
    dr = wred(dr) * s;   // logit for rot (cand index 1)
    df = wred(df) * s;   // logit for ref (cand index 0)
    const float mlg = fmaxf(dr, df);
    const float er = __expf(dr - mlg), ef = __expf(df - mlg);
    const float inv_s = 1.f / (er + ef);
    const float w_rot = er * inv_s, w_ref = ef * inv_s;

    float2 attq[8], rl[8];
    float un2 = 0.f, rn2 = 0.f;
    #pragma unroll
    for (int j = 0; j < 8; j++) {
        attq[j].x = w_ref * refq[j].x + w_rot * rotq[j].x;
        attq[j].y = w_ref * refq[j].y + w_rot * rotq[j].y;
        rl[j] = *(const float2*)(relp + j * 64 + lane * 2);
        un2 += attq[j].x * attq[j].x + attq[j].y * attq[j].y;
        rn2 += rl[j].x * rl[j].x + rl[j].y * rl[j].y;
    }
    un2 = wred(un2); rn2 = wred(rn2);

    const float cv = curv[row];
    const float c  = (cv > 20.f) ? cv : log1pf(__expf(cv));  // softplus
    const float sc = sqrtf(c);
    const float un = fmaxf(sqrtf(un2), MIN_NORM);
    const float rn = fmaxf(sqrtf(rn2), MIN_NORM);
    const float fu = tanh_clip(sc * un) / (sc * un);   // expmap0 scales
    const float fr = tanh_clip(sc * rn) / (sc * rn);

    float x2 = 0.f, y2 = 0.f, xy = 0.f;
    #pragma unroll
    for (int j = 0; j < 8; j++) {
        attq[j].x *= fu; attq[j].y *= fu;   // lhs
        rl[j].x   *= fr; rl[j].y   *= fr;   // rel_h
        x2 += attq[j].x * attq[j].x + attq[j].y * attq[j].y;
        y2 += rl[j].x * rl[j].x + rl[j].y * rl[j].y;
        xy += attq[j].x * rl[j].x + attq[j].y * rl[j].y;
    }
    x2 = wred(x2); y2 = wred(y2); xy = wred(xy);

    const float k1 = 1.f + 2.f * c * xy + c * y2;
    const float k2 = 1.f - c * x2;
    const float dn = 1.f + 2.f * c * xy + c * c * x2 * y2;
    const float invdn = 1.f / fmaxf(dn, MIN_NORM);

    float2 rv[8]; float n2 = 0.f;
    #pragma unroll
    for (int j = 0; j < 8; j++) {
        rv[j].x = (k1 * attq[j].x + k2 * rl[j].x) * invdn;
        rv[j].y = (k1 * attq[j].y + k2 * rl[j].y) * invdn;
        n2 += rv[j].x * rv[j].x + rv[j].y * rv[j].y;
    }
    n2 = wred(n2);
    const float norm = fmaxf(sqrtf(n2), MIN_NORM);
    const float maxn = (1.f - BALL_EPS) / sc;
    float fsc = 1.f, x2f = n2;
    if (norm > maxn) { fsc = maxn / norm; x2f = maxn * maxn; }

    float* rp = res + (size_t)row * 512;
    #pragma unroll
    for (int j = 0; j < 8; j++) {
        float2 o; o.x = rv[j].x * fsc; o.y = rv[j].y * fsc;
        *(float2*)(rp + j * 64 + lane * 2) = o;
    }
    if (lane == 0) rowpack[row] = make_float4(c, sc, x2f, head_bias[row]);
}

// ---------------------------------------------------------------------------
// Kernel 2: tail norms. One wave32 per tail row. colpack = {||v||, tail_bias}.
// ---------------------------------------------------------------------------
__global__ __launch_bounds__(256) void atth_tailprep(
    const float* __restrict__ tail, const float* __restrict__ tail_bias,
    float2* __restrict__ colpack)
{
    const int lane = threadIdx.x & 31;
    const int row  = blockIdx.x * 8 + (threadIdx.x >> 5);
    const float* tp = tail + (size_t)row * 512;
    float s2 = 0.f;
    #pragma unroll
    for (int j = 0; j < 4; j++) {
        float4 v = *(const float4*)(tp + j * 128 + lane * 4);
        s2 += v.x * v.x + v.y * v.y + v.z * v.z + v.w * v.w;
    }
    s2 = wred(s2);
    if (lane == 0) colpack[row] = make_float2(sqrtf(s2), tail_bias[row]);
}

// ---------------------------------------------------------------------------
// Kernel 3: batched GEMM (f32 WMMA 16x16x4) + hyperbolic-distance epilogue.
// Block: 256 threads = 8 waves, tile 64(M) x 128(N); wave = 32x32 via 2x2
// accumulators. K staged through double-buffered LDS filled with CDNA5
// GLOBAL_LOAD_ASYNC_TO_LDS_B128 (ASYNCcnt), fully overlapped with WMMA.
// LDS stride 36 floats: conflict-free ds_load_b64 frag reads, 16B-aligned fill.
// ---------------------------------------------------------------------------
#define WMMA_F32X4(a, b, c) \
    __builtin_amdgcn_wmma_f32_16x16x4_f32(false, (a), false, (b), (short)0, (c), false, false)

__global__ __launch_bounds__(256) void atth_score(
    const float* __restrict__ res, const float4* __restrict__ rowpack,
    const float2* __restrict__ colpack, const float* __restrict__ tail,
    float* __restrict__ out)
{
    constexpr int LDA = 36;
    __shared__ float Alds[2][64 * LDA];
    __shared__ float Blds[2][128 * LDA];

    const int tid  = threadIdx.x;
    const int lane = tid & 31;
    const int wave = tid >> 5;       // 0..7
    const int wm   = wave >> 2;      // 0..1 -> 32-row strip
    const int wn   = wave & 3;       // 0..3 -> 32-col strip
    const int ln   = lane & 15;
    const int hf   = lane >> 4;      // K-half (A/B frag) / M-half (C frag)
    const int koff = hf * 2;

    const int chunk  = blockIdx.z;
    const int m_base = blockIdx.y * 64;
    const int n_base = blockIdx.x * 128;
    const float* Ap = res  + ((size_t)chunk * 1024 + m_base) * 512;
    const float* Bp = tail + ((size_t)chunk * 1024 + n_base) * 512;

    v8f acc[2][2] = {};

    const int lr = tid >> 3;         // 0..31: fill row within pass
    const int lc = (tid & 7) * 4;    // 0..28: fill col (float4)

    // Per-thread LDS byte addresses for the 6 async b128 fills, per buffer.
    unsigned ldsA[2][2], ldsB[2][4];
    #pragma unroll
    for (int b = 0; b < 2; b++) {
        #pragma unroll
        for (int rr = 0; rr < 2; rr++)
            ldsA[b][rr] = lds_off(&Alds[b][(lr + rr * 32) * LDA + lc]);
        #pragma unroll
        for (int rr = 0; rr < 4; rr++)
            ldsB[b][rr] = lds_off(&Blds[b][(lr + rr * 32) * LDA + lc]);
    }

    auto prefetch = [&](int buf, int k0) {
        #pragma unroll
        for (int rr = 0; rr < 2; rr++)
            async_copy_b128(ldsA[buf][rr], Ap, (unsigned)(((lr + rr * 32) * 512 + k0 + lc) * 4));
        #pragma unroll
        for (int rr = 0; rr < 4; rr++)
            async_copy_b128(ldsB[buf][rr], Bp, (unsigned)(((lr + rr * 32) * 512 + k0 + lc) * 4));
    };

    prefetch(0, 0);

    for (int k0 = 0; k0 < 512; k0 += 32) {
        const int buf = (k0 >> 5) & 1;
        wait_async0();        // own async fills of `buf` complete
        __syncthreads();      // everyone's fills complete; prior compute retired
        if (k0 + 32 < 512) prefetch(buf ^ 1, k0 + 32);

        const float* Abuf = &Alds[buf][0];
        const float* Bbuf = &Blds[buf][0];
        #pragma unroll
        for (int kk = 0; kk < 32; kk += 4) {
            // A frag (16x4 f32): lanes 0-15 hold K={0,1}, lanes 16-31 K={2,3}
            v2f a0 = *(const v2f*)&Abuf[(wm * 32 +      ln) * LDA + kk + koff];
            v2f a1 = *(const v2f*)&Abuf[(wm * 32 + 16 + ln) * LDA + kk + koff];
            // B frag (4x16 f32): B[k][n] = tail[n][k]
            v2f b0 = *(const v2f*)&Bbuf[(wn * 32 +      ln) * LDA + kk + koff];
            v2f b1 = *(const v2f*)&Bbuf[(wn * 32 + 16 + ln) * LDA + kk + koff];
            acc[0][0] = WMMA_F32X4(a0, b0, acc[0][0]);
            acc[0][1] = WMMA_F32X4(a0, b1, acc[0][1]);
            acc[1][0] = WMMA_F32X4(a1, b0, acc[1][0]);
            acc[1][1] = WMMA_F32X4(a1, b1, acc[1][1]);
        }
        __syncthreads();      // compute on `buf` retired before it is refilled
    }

    // Epilogue. C/D layout: VGPR e -> M = e (lanes 0-15) / e+8 (lanes 16-31), N = ln.
    // Hoisted scalars: 2 colpack (b64) + 16 rowpack (b128) loads per lane.
    float2 cp[2];
    #pragma unroll
    for (int j = 0; j < 2; j++)
        cp[j] = colpack[(size_t)chunk * 1024 + n_base + wn * 32 + j * 16 + ln];

    #pragma unroll
    for (int i = 0; i < 2; i++) {
        #pragma unroll
        for (int e = 0; e < 8; e++) {
            const int row = m_base + wm * 32 + i * 16 + e + hf * 8;
            const float4 rp = rowpack[(size_t)chunk * 1024 + row];
            const float cc = rp.x, sc = rp.y, x2 = rp.z, hb = rp.w;
            #pragma unroll
            for (int j = 0; j < 2; j++) {
                const float vn = cp[j].x, tb = cp[j].y;
                const float dot = acc[i][j][e];
                const float xv = dot / vn;
                const float g  = tanh_clip(sc * vn) / sc;
                const float cg = cc * g;
                const float c1 = 1.f - 2.f * cg * xv + cg * g;
                const float c2 = 1.f - cc * x2;
                const float num2 = c1 * c1 * x2 + c2 * c2 * g * g - 2.f * c1 * c2 * g * xv;
                const float num = sqrtf(fmaxf(num2, 0.f));
                const float denom = 1.f - 2.f * cg * xv + cg * cg * x2;
                float z = sc * num / fmaxf(denom, MIN_NORM);
                z = fminf(fmaxf(z, -1.f + 1e-5f), 1.f - 1e-5f);
                const float at = 0.5f * __logf((1.f + z) / (1.f - z));
                const float dist = 2.f * at / sc;
                const int col = n_base + wn * 32 + j * 16 + ln;
                out[((size_t)chunk * 1024 + row) * 1024 + col] = -dist * dist + hb + tb;
            }
        }
    }
}

extern "C" void kernel_launch(void* const* d_in, const int* in_sizes, int n_in,
                              void* d_out, int out_size, void* d_ws, size_t ws_size,
                              hipStream_t stream) {
    const float* head      = (const float*)d_in[0];
    const float* head_bias = (const float*)d_in[1];
    const float* rel       = (const float*)d_in[2];
    const float* rel_diag  = (const float*)d_in[3];
    const float* curv      = (const float*)d_in[4];
    const float* context   = (const float*)d_in[5];
    const float* scale     = (const float*)d_in[6];
    const float* tail      = (const float*)d_in[7];
    const float* tail_bias = (const float*)d_in[8];
    float* out = (float*)d_out;

    // nc=16, cs=1024, ns=1024, d=512 (from setup_inputs)
    constexpr int B = 16384;            // nc*cs == nc*ns
    // workspace layout: res (B*512 f32) | rowpack (B float4) | colpack (B float2)
    float*  res     = (float*)d_ws;
    float4* rowpack = (float4*)((char*)d_ws + (size_t)B * 512 * sizeof(float));
    float2* colpack = (float2*)((char*)rowpack + (size_t)B * sizeof(float4));

    atth_prep<<<B / 8, 256, 0, stream>>>(head, head_bias, rel, rel_diag, curv,
                                         context, scale, res, rowpack);
    atth_tailprep<<<B / 8, 256, 0, stream>>>(tail, tail_bias, colpack);

    dim3 grid(1024 / 128, 1024 / 64, 16);   // (ns tiles, cs tiles, nc)
    atth_score<<<grid, 256, 0, stream>>>(res, rowpack, colpack, tail, out);
}